// PrimalNN_41094247088344
// MI455X (gfx1250) — compile-verified
//
#include <hip/hip_runtime.h>

// ---------------------------------------------------------------------------
// PrimalNN forward on MI455X (gfx1250), f32 WMMA path + async global->LDS.
//
//   out    = W4·relu(W3·relu(W2·relu(W1·b + b1) + b2) + b3) + b4
//   Bias   = b · WbProj^T
//   z      = out;  repeat 10x:  z = mask(Bias + z · WzProj^T)
//   d_out  = [ z_star | out ]   (each 64x512 f32)
//
// Jacobian is discarded by the reference -> not computed.
// Convergence residual max|z·A^T - b| can never reach 1e-8 (fixed point of the
// projection is unrelated to A·z=b), so the loop always runs MAX_ITER=10 times
// -> A is unused and no residual GEMM is launched.
// ---------------------------------------------------------------------------

#define FREE_NUM 100

typedef __attribute__((ext_vector_type(2))) float v2f;
typedef __attribute__((ext_vector_type(8))) float v8f;

#define EPI_NONE 0
#define EPI_RELU 1
#define EPI_PROJ 2

// GLOBAL_LOAD_ASYNC_TO_LDS_B128 (GV mode): per-lane 16B memory->LDS copy,
// tracked by ASYNCcnt, no VGPR data round trip (CDNA5 ISA 15.18.3 / 08 §4).
__device__ __forceinline__ void async_ld_b128(const float* gaddr, unsigned lds_off)
{
    asm volatile("global_load_async_to_lds_b128 %0, %1, off"
                 :: "v"(lds_off), "v"(gaddr) : "memory");
}
__device__ __forceinline__ void wait_async_le10()
{   // previous chunk's 10 per-wave async issues retired (in-order completion)
    asm volatile("s_wait_asynccnt 0xA" ::: "memory");
}
__device__ __forceinline__ void wait_async_le0()
{
    asm volatile("s_wait_asynccnt 0x0" ::: "memory");
}

// D = epilogue( X @ W^T [+ bias] ), X:[M,K] row-major, W:[N,K] row-major
// (nn.Linear convention -> "NT" GEMM, both operands K-contiguous).
// Block: 128 threads = 4 wave32; block tile 16(M) x 64(N); wave tile 16x16.
// Grid: dim3(N/64, M/16). Requires K % 64 == 0 (448/512/1024 all qualify).
template <int EPI, bool HAS_BIAS>
__global__ __launch_bounds__(128)
void gemm_nt_wmma_f32(const float* __restrict__ X, const float* __restrict__ W,
                      const float* __restrict__ bias,
                      const float* __restrict__ bias2d,
                      float* __restrict__ D, int M, int N, int K)
{
    constexpr int LDX = 68;            // padded LDS stride (floats): 272B row
                                       // -> 16B-aligned b128 stage, fragment
                                       // reads hit banks (4m+k) mod 64
    __shared__ float Xs[2][16 * LDX];  // double-buffered 16 x 64 X tile
    __shared__ float Ws[2][64 * LDX];  // double-buffered 64 x 64 W tile

    const int tid  = threadIdx.x;
    const int wave = tid >> 5;
    const int lane = tid & 31;
    const int hh   = lane >> 4;        // half-wave: selects K pair {0,1}/{2,3}
    const int ml   = lane & 15;        // M index (A) / N index (B,C,D)

    const int m0 = blockIdx.y * 16;
    const int n0 = blockIdx.x * 64;

    // Issue one chunk's staging: 2 (X) + 8 (W) async b128 per wave.
    auto stage = [&](int c, int buf) {
        const int kc = c * 64;
        for (int i = tid; i < 16 * 16; i += 128) {       // 2 iters, no diverge
            const int r = i >> 4, c4 = i & 15;
            async_ld_b128(X + (size_t)(m0 + r) * K + kc + c4 * 4,
                          (unsigned)(uintptr_t)&Xs[buf][r * LDX + c4 * 4]);
        }
        for (int i = tid; i < 64 * 16; i += 128) {       // 8 iters, no diverge
            const int r = i >> 4, c4 = i & 15;
            async_ld_b128(W + (size_t)(n0 + r) * K + kc + c4 * 4,
                          (unsigned)(uintptr_t)&Ws[buf][r * LDX + c4 * 4]);
        }
    };

    v8f acc = {};
    const int nch = K / 64;

    stage(0, 0);
    for (int c = 0; c < nch; ++c) {
        if (c + 1 < nch) {                 // prefetch next chunk behind compute
            stage(c + 1, (c + 1) & 1);
            wait_async_le10();             // chunk c landed (this wave's part)
        } else {
            wait_async_le0();
        }
        __syncthreads();                   // chunk c visible to all waves

        // Fragment layout per ISA 16x4 f32 A-matrix table:
        //   lanes 0-15 : v0=K0, v1=K1 ; lanes 16-31 : v0=K2, v1=K3
        // B (4x16) mirrors it with N striped across lanes.
        const float* xrow = &Xs[c & 1][ml * LDX];
        const float* wrow = &Ws[c & 1][(wave * 16 + ml) * LDX];
        #pragma unroll
        for (int kk = 0; kk < 64; kk += 4) {
            v2f af = *(const v2f*)(xrow + kk + 2 * hh);
            v2f bf = *(const v2f*)(wrow + kk + 2 * hh);
            // (neg_a, A, neg_b, B, c_mod, C, reuse_a, reuse_b)
            acc = __builtin_amdgcn_wmma_f32_16x16x4_f32(
                false, af, false, bf, (short)0, acc, false, false);
        }
        __syncthreads();                   // buffer c&1 free for reuse at c+1
    }

    // C/D layout: VGPR r, lanes 0-15 -> (M=r, N=lane); lanes 16-31 -> (M=r+8).
    const int n = n0 + wave * 16 + ml;
    const float bv = HAS_BIAS ? bias[n] : 0.0f;
    #pragma unroll
    for (int r = 0; r < 8; ++r) {
        const int m = m0 + r + 8 * hh;
        float v;
        if (EPI == EPI_PROJ) {
            v = acc[r] + bias2d[(size_t)m * N + n];
            if (v < 0.0f && n >= FREE_NUM) v = 0.0f;   // partial ReLU block
        } else {
            v = acc[r] + bv;
            if (EPI == EPI_RELU) v = fmaxf(v, 0.0f);
        }
        D[(size_t)m * N + n] = v;
    }
}

extern "C" void kernel_launch(void* const* d_in, const int* in_sizes, int n_in,
                              void* d_out, int out_size, void* d_ws, size_t ws_size,
                              hipStream_t stream)
{
    (void)in_sizes; (void)n_in; (void)out_size; (void)ws_size;

    const float* b      = (const float*)d_in[0];   // [64,448]
    const float* W1     = (const float*)d_in[1];   // [1024,448]
    const float* b1     = (const float*)d_in[2];   // [1024]
    const float* W2     = (const float*)d_in[3];   // [1024,1024]
    const float* b2     = (const float*)d_in[4];
    const float* W3     = (const float*)d_in[5];   // [1024,1024]
    const float* b3     = (const float*)d_in[6];
    const float* W4     = (const float*)d_in[7];   // [512,1024]
    const float* b4     = (const float*)d_in[8];
    /* d_in[9] = A [448,512]: only feeds the convergence test, which can never
       fire (see header comment) -> unused. */
    const float* WzProj = (const float*)d_in[10];  // [512,512]
    const float* WbProj = (const float*)d_in[11];  // [512,448]

    float* zstar = (float*)d_out;                  // [64,512]
    float* outp  = zstar + 64 * 512;               // [64,512]

    // Workspace layout (f32 elements), ~0.9 MB total:
    float* h1   = (float*)d_ws;                    // 64*1024
    float* h2   = h1 + 64 * 1024;                  // 64*1024
    float* Bias = h2 + 64 * 1024;                  // 64*512
    float* zA   = Bias + 64 * 512;                 // 64*512
    float* zB   = zA + 64 * 512;                   // 64*512

    const dim3 blk(128);
    const dim3 gN1024(1024 / 64, 64 / 16);
    const dim3 gN512(512 / 64, 64 / 16);

    // MLP trunk (stream-ordered -> layer dependencies honored)
    gemm_nt_wmma_f32<EPI_RELU, true ><<<gN1024, blk, 0, stream>>>(b,  W1, b1, nullptr, h1,   64, 1024, 448);
    gemm_nt_wmma_f32<EPI_RELU, true ><<<gN1024, blk, 0, stream>>>(h1, W2, b2, nullptr, h2,   64, 1024, 1024);
    gemm_nt_wmma_f32<EPI_RELU, true ><<<gN1024, blk, 0, stream>>>(h2, W3, b3, nullptr, h1,   64, 1024, 1024);
    gemm_nt_wmma_f32<EPI_NONE, true ><<<gN512,  blk, 0, stream>>>(h1, W4, b4, nullptr, outp, 64, 512,  1024);

    // Bias = b @ WbProj^T
    gemm_nt_wmma_f32<EPI_NONE, false><<<gN512,  blk, 0, stream>>>(b, WbProj, nullptr, nullptr, Bias, 64, 512, 448);

    // Projection: exactly 10 iterations (convergence test provably inert).
    // Ping-pong zA/zB; iter 0 reads `out` in place, iter 9 writes z_star.
    const float* cur = outp;
    for (int t = 0; t < 10; ++t) {
        float* dst = (t == 9) ? zstar : ((t & 1) ? zB : zA);
        gemm_nt_wmma_f32<EPI_PROJ, false><<<gN512, blk, 0, stream>>>(cur, WzProj, nullptr, Bias, dst, 64, 512, 512);
        cur = dst;
    }
}